// DetectionLayer_84095459655722
// MI455X (gfx1250) — compile-verified
//
#include <hip/hip_runtime.h>
#include <stdint.h>

#define BDIM 32
#define NBOX 30000
#define NUM_CLASSES 81
#define MAX_DET 100
#define MIN_CONF 0.7f
#define IOU_THR 0.3f
#define BLK 1024

// ---------------- CDNA5 async-to-LDS helpers (builtin with asm fallback) ----
#if defined(__has_builtin)
#if __has_builtin(__builtin_amdgcn_global_load_async_to_lds_b128)
#define HAVE_ASYNC_B128 1
#endif
#if __has_builtin(__builtin_amdgcn_s_wait_asynccnt)
#define HAVE_WAIT_ASYNC 1
#endif
#endif

typedef int v4i __attribute__((vector_size(16)));
typedef __attribute__((address_space(1))) v4i* gp128_t;
typedef __attribute__((address_space(3))) v4i* lp128_t;

__device__ __forceinline__ void async_copy16(const float* g, float* l) {
#ifdef HAVE_ASYNC_B128
  __builtin_amdgcn_global_load_async_to_lds_b128(
      (gp128_t)(uintptr_t)g,   // global src (AS1, int4*)
      (lp128_t)l,              // LDS dst (addrspacecast generic->AS3)
      0, 0);
#else
  // async VMEM path per CDNA5 ISA:
  // global_load_async_to_lds_b128 vdst(lds addr), vaddr(64b), off
  asm volatile("global_load_async_to_lds_b128 %0, %1, off"
               :
               : "v"((uint32_t)(uintptr_t)l), "v"(g)
               : "memory");
#endif
}

__device__ __forceinline__ void wait_async0() {
#ifdef HAVE_WAIT_ASYNC
  __builtin_amdgcn_s_wait_asynccnt(0);
#else
  asm volatile("s_wait_asynccnt 0" ::: "memory");
#endif
}

// Order-preserving float->uint key (total order, works for negatives).
__device__ __forceinline__ uint32_t fkey(float s) {
  uint32_t k = __float_as_uint(s);
  return (k & 0x80000000u) ? ~k : (k | 0x80000000u);
}
__device__ __forceinline__ unsigned long long packsi(float s, int i) {
  return ((unsigned long long)fkey(s) << 32) | (uint32_t)(~i);  // ~i: ties -> smallest idx
}

// ---------------- Kernel 1: decode + clip + mask (memory bound) -------------
__global__ __launch_bounds__(256) void det_decode_kernel(
    const float* __restrict__ rois, const float* __restrict__ cls,
    const float* __restrict__ window, float* __restrict__ boxes,
    float* __restrict__ mscore, int* __restrict__ clsid) {
  int t = blockIdx.x * blockDim.x + threadIdx.x;
  if (t >= BDIM * NBOX) return;
  int b = t / NBOX;

  float4 r = ((const float4*)rois)[t];
  const float* c = cls + (size_t)t * 6;
  float dy = c[0] * 0.1f, dx = c[1] * 0.1f;
  float dh = c[2] * 0.2f, dw = c[3] * 0.2f;
  int cid = (int)c[4];
  float sc = c[5];

  float h = r.z - r.x, w = r.w - r.y;
  float cy = r.x + 0.5f * h + dy * h;
  float cx = r.y + 0.5f * w + dx * w;
  h = h * expf(dh);
  w = w * expf(dw);
  float y1 = cy - 0.5f * h, x1 = cx - 0.5f * w;
  float y2 = cy + 0.5f * h, x2 = cx + 0.5f * w;

  float4 win = ((const float4*)window)[b];
  y1 = fminf(fmaxf(y1, win.x), win.z);
  y2 = fminf(fmaxf(y2, win.x), win.z);
  x1 = fminf(fmaxf(x1, win.y), win.w);
  x2 = fminf(fmaxf(x2, win.y), win.w);

  ((float4*)boxes)[t] = make_float4(y1, x1, y2, x2);
  bool valid = (cid > 0) && (sc >= MIN_CONF);
  mscore[t] = valid ? sc : -1.0f;
  clsid[t] = cid;
}

// ---------------- Kernel 2: greedy NMS, one workgroup per image -------------
__global__ __launch_bounds__(BLK) void det_nms_kernel(
    const float* __restrict__ boxes, const float* __restrict__ mscore,
    const int* __restrict__ clsid, float* __restrict__ out) {
  __shared__ float s_sc[NBOX];                // 120 KB
  __shared__ unsigned char s_cl[NBOX];        // 30 KB
  __shared__ unsigned long long s_red[BLK / 32];
  __shared__ float s_best[4];

  const int b = blockIdx.x;
  const int tid = threadIdx.x;
  const int wid = tid >> 5, lane = tid & 31;
  const float* bx = boxes + (size_t)b * NBOX * 4;
  const float* ms = mscore + (size_t)b * NBOX;
  const int* ci = clsid + (size_t)b * NBOX;
  float* orow = out + (size_t)b * MAX_DET * 6;

  // Stage masked scores into LDS with async-to-LDS b128 DMA (NBOX % 4 == 0).
  for (int i4 = tid; i4 < NBOX / 4; i4 += BLK)
    async_copy16(ms + i4 * 4, &s_sc[i4 * 4]);
  // Stage class ids (needs a transform, plain path).
  for (int i = tid; i < NBOX; i += BLK) s_cl[i] = (unsigned char)ci[i];
  wait_async0();
  __syncthreads();

  // Initial per-thread argmax over its strided chunk.
  unsigned long long lp = 0ull;
  for (int i = tid; i < NBOX; i += BLK) {
    float s = s_sc[i];
    if (s >= 0.0f) {
      unsigned long long p = packsi(s, i);
      lp = (p > lp) ? p : lp;
    }
  }

  int it = 0;
  for (; it < MAX_DET; ++it) {
    // Wave reduce (wave32), then 32-partial LDS reduce, redundantly on all threads.
    unsigned long long v = lp;
    #pragma unroll
    for (int o = 16; o > 0; o >>= 1) {
      unsigned long long u = __shfl_xor(v, o, 32);
      v = (u > v) ? u : v;
    }
    if (lane == 0) s_red[wid] = v;
    __syncthreads();
    unsigned long long bp = s_red[0];
    #pragma unroll
    for (int k = 1; k < BLK / 32; ++k) {
      unsigned long long u = s_red[k];
      bp = (u > bp) ? u : bp;
    }
    uint32_t key = (uint32_t)(bp >> 32);
    uint32_t bidx = ~(uint32_t)bp;
    uint32_t bbits = (key & 0x80000000u) ? (key ^ 0x80000000u) : ~key;
    float best = __uint_as_float(bbits);  // bp==0 -> NaN -> invalid
    if (!(best > 0.0f)) break;

    // Broadcast the winning box through LDS via async DMA (single lane).
    if (tid == 0) {
      async_copy16(bx + (size_t)bidx * 4, s_best);
      wait_async0();
    }
    __syncthreads();  // also protects s_red / s_best reuse across iterations

    float by1 = s_best[0], bx1 = s_best[1], by2 = s_best[2], bx2 = s_best[3];
    int bcid = s_cl[bidx];
    float barea = (by2 - by1) * (bx2 - bx1);

    // Fused suppression + next-iteration local argmax (thread-local chunk).
    lp = 0ull;
    for (int i = tid; i < NBOX; i += BLK) {
      float s = s_sc[i];
      if (s < 0.0f) continue;
      if ((uint32_t)i == bidx) { s_sc[i] = -1.0f; continue; }
      float4 q = ((const float4*)bx)[i];
      float yy1 = fmaxf(by1, q.x), xx1 = fmaxf(bx1, q.y);
      float yy2 = fminf(by2, q.z), xx2 = fminf(bx2, q.w);
      float inter = fmaxf(yy2 - yy1, 0.0f) * fmaxf(xx2 - xx1, 0.0f);
      float area = (q.z - q.x) * (q.w - q.y);
      float iou = inter / (barea + area - inter + 1e-8f);
      if (iou > IOU_THR && (int)s_cl[i] == bcid) {
        s_sc[i] = -1.0f;
      } else {
        unsigned long long p = packsi(s, i);
        lp = (p > lp) ? p : lp;
      }
    }

    if (tid == 0) {
      orow[it * 6 + 0] = by1;
      orow[it * 6 + 1] = bx1;
      orow[it * 6 + 2] = by2;
      orow[it * 6 + 3] = bx2;
      orow[it * 6 + 4] = (float)bcid;
      orow[it * 6 + 5] = best;
    }
  }
  // Zero-fill the remaining rows (disjoint from written rows; deterministic).
  for (int k = it * 6 + tid; k < MAX_DET * 6; k += BLK) orow[k] = 0.0f;
}

// ---------------- Host launch ----------------------------------------------
extern "C" void kernel_launch(void* const* d_in, const int* in_sizes, int n_in,
                              void* d_out, int out_size, void* d_ws, size_t ws_size,
                              hipStream_t stream) {
  const float* rois = (const float*)d_in[0];            // [B,N,4]
  const float* classifications = (const float*)d_in[1]; // [B,N,6]
  const float* window = (const float*)d_in[2];          // [B,4]
  float* out = (float*)d_out;                           // [B,MAX_DET,6]

  // Workspace layout: boxes [B*N*4] f32 | mscore [B*N] f32 | clsid [B*N] i32
  float* boxes = (float*)d_ws;
  float* mscore = boxes + (size_t)BDIM * NBOX * 4;
  int* clsid = (int*)(mscore + (size_t)BDIM * NBOX);

  int total = BDIM * NBOX;
  det_decode_kernel<<<(total + 255) / 256, 256, 0, stream>>>(
      rois, classifications, window, boxes, mscore, clsid);
  det_nms_kernel<<<BDIM, BLK, 0, stream>>>(boxes, mscore, clsid, out);
}